// MeanAggregatorSparse_54863912239169
// MI455X (gfx1250) — compile-verified
//
#include <hip/hip_runtime.h>

typedef __attribute__((ext_vector_type(2))) float v2f;
typedef __attribute__((ext_vector_type(4))) float v4f;
typedef __attribute__((ext_vector_type(8))) float v8f;

#define D_FEAT   128
#define OUT_DIM  128
#define K_TOT    256          // concat(self, agg)
#define LDS_PITCH 260         // 260 % 64 == 4 -> conflict-free b64 lane pattern

// ---------------------------------------------------------------------------
// Kernel 0: zero the workspace (agg sums + counts) every call (deterministic).
// ---------------------------------------------------------------------------
__global__ void zero_f32(float* __restrict__ p, long long n) {
  long long i = (long long)blockIdx.x * blockDim.x + threadIdx.x;
  if (i < n) p[i] = 0.0f;
}

// ---------------------------------------------------------------------------
// Kernel 1: edge scatter-add. One wave (32 lanes) per edge; each lane owns a
// float4 chunk of the 128-wide feature row. nbr_feat is streamed exactly once
// -> non-temporal loads keep L2 free for the 25.6MB accumulator.
// ---------------------------------------------------------------------------
__global__ __launch_bounds__(256) void edge_scatter(
    const float* __restrict__ nbr, const int* __restrict__ src,
    float* __restrict__ aggsum, float* __restrict__ cnt, int n_edges) {
  int edge = blockIdx.x * 8 + (threadIdx.x >> 5);
  if (edge >= n_edges) return;
  int lane = threadIdx.x & 31;
  int s = src[edge];                       // wave-uniform -> scalarized
  const v4f* row = (const v4f*)(nbr + (size_t)edge * D_FEAT);
  v4f v = __builtin_nontemporal_load(row + lane);
  float* dst = aggsum + (size_t)s * D_FEAT + lane * 4;
  atomicAdd(dst + 0, v[0]);
  atomicAdd(dst + 1, v[1]);
  atomicAdd(dst + 2, v[2]);
  atomicAdd(dst + 3, v[3]);
  if (lane == 0) atomicAdd(cnt + s, 1.0f);
}

// ---------------------------------------------------------------------------
// Kernel 2: fused mean + concat + GEMM with V_WMMA_F32_16X16X4_F32.
// 256 threads = 8 waves; each wave computes one 16-node x 128-out row block.
// W (256x128 f32 = 128KB) is staged transposed into LDS once per block.
// B-fragments are batch-loaded (8 at a time) so one s_wait_dscnt covers all
// eight DS loads instead of one wait per WMMA; per-tile LDS base pointers are
// hoisted out of the K loop so the inner loop only advances by kk.
// ---------------------------------------------------------------------------
__global__ __launch_bounds__(256) void fused_mean_gemm(
    const float* __restrict__ self_feat,
    const float* __restrict__ aggsum,
    const float* __restrict__ cnt,
    const float* __restrict__ W,
    float* __restrict__ out,
    int n_nodes, int n_row_blocks) {
  extern __shared__ float sW[];  // sW[n * LDS_PITCH + k] = W[k][n]

  // Cooperative transpose-stage of W into LDS (one time per block).
  for (int i = threadIdx.x; i < K_TOT * OUT_DIM; i += blockDim.x) {
    int k = i >> 7;             // 0..255
    int n = i & (OUT_DIM - 1);  // 0..127
    sW[n * LDS_PITCH + k] = W[i];
  }
  __syncthreads();

  int wave = threadIdx.x >> 5;
  int lane = threadIdx.x & 31;
  int rb = blockIdx.x * 8 + wave;
  if (rb >= n_row_blocks) return;          // wave-uniform exit (EXEC stays full)

  int node_base = rb * 16;
  int r = lane & 15;                        // M row within 16-tile
  int g = lane >> 4;                        // K half-select (ISA A16x4 layout)
  int node = node_base + r;
  if (node >= n_nodes) node = n_nodes - 1;  // clamp loads; stores guarded below

  float c   = cnt[node];
  float inv = (c > 0.0f) ? (1.0f / c) : 0.0f;

  const float* A0 = self_feat + (size_t)node * D_FEAT + 2 * g;  // k in [0,128)
  const float* A1 = aggsum    + (size_t)node * D_FEAT + 2 * g;  // k in [128,256)

  // Per-tile LDS base pointers (col = t*16 + r, k origin = 2*g), hoisted.
  const float* bp0 = &sW[(0 * 16 + r) * LDS_PITCH + 2 * g];
  const float* bp1 = &sW[(1 * 16 + r) * LDS_PITCH + 2 * g];
  const float* bp2 = &sW[(2 * 16 + r) * LDS_PITCH + 2 * g];
  const float* bp3 = &sW[(3 * 16 + r) * LDS_PITCH + 2 * g];
  const float* bp4 = &sW[(4 * 16 + r) * LDS_PITCH + 2 * g];
  const float* bp5 = &sW[(5 * 16 + r) * LDS_PITCH + 2 * g];
  const float* bp6 = &sW[(6 * 16 + r) * LDS_PITCH + 2 * g];
  const float* bp7 = &sW[(7 * 16 + r) * LDS_PITCH + 2 * g];

  v8f acc[8] = {};   // 8 N-tiles of 16 -> full 16x128 output row block

  // ---- K phase 1: self features ----
  #pragma unroll 4
  for (int kk = 0; kk < D_FEAT; kk += 4) {
    v2f a = *(const v2f*)(A0 + kk);
    v2f b[8];                               // batch all 8 B-frags, then WMMA
    b[0] = *(const v2f*)(bp0 + kk);
    b[1] = *(const v2f*)(bp1 + kk);
    b[2] = *(const v2f*)(bp2 + kk);
    b[3] = *(const v2f*)(bp3 + kk);
    b[4] = *(const v2f*)(bp4 + kk);
    b[5] = *(const v2f*)(bp5 + kk);
    b[6] = *(const v2f*)(bp6 + kk);
    b[7] = *(const v2f*)(bp7 + kk);
    #pragma unroll
    for (int t = 0; t < 8; ++t)
      acc[t] = __builtin_amdgcn_wmma_f32_16x16x4_f32(
          false, a, false, b[t], (short)0, acc[t], false, false);
  }
  // ---- K phase 2: aggregated neighbor mean (sum * inv) ----
  #pragma unroll 4
  for (int kk = 0; kk < D_FEAT; kk += 4) {
    v2f a = *(const v2f*)(A1 + kk);
    a[0] *= inv; a[1] *= inv;
    v2f b[8];
    b[0] = *(const v2f*)(bp0 + D_FEAT + kk);
    b[1] = *(const v2f*)(bp1 + D_FEAT + kk);
    b[2] = *(const v2f*)(bp2 + D_FEAT + kk);
    b[3] = *(const v2f*)(bp3 + D_FEAT + kk);
    b[4] = *(const v2f*)(bp4 + D_FEAT + kk);
    b[5] = *(const v2f*)(bp5 + D_FEAT + kk);
    b[6] = *(const v2f*)(bp6 + D_FEAT + kk);
    b[7] = *(const v2f*)(bp7 + D_FEAT + kk);
    #pragma unroll
    for (int t = 0; t < 8; ++t)
      acc[t] = __builtin_amdgcn_wmma_f32_16x16x4_f32(
          false, a, false, b[t], (short)0, acc[t], false, false);
  }

  // C/D layout: VGPR j -> M = j + 8*g, N = (lane&15) + 16*t. Stream out (NT).
  int mbase = node_base + g * 8;
  #pragma unroll
  for (int t = 0; t < 8; ++t) {
    #pragma unroll
    for (int j = 0; j < 8; ++j) {
      int m = mbase + j;
      if (m < n_nodes)
        __builtin_nontemporal_store(
            acc[t][j], out + (size_t)m * OUT_DIM + t * 16 + r);
    }
  }
}

// ---------------------------------------------------------------------------
extern "C" void kernel_launch(void* const* d_in, const int* in_sizes, int n_in,
                              void* d_out, int out_size, void* d_ws, size_t ws_size,
                              hipStream_t stream) {
  const float* self_feat = (const float*)d_in[0];
  const float* nbr_feat  = (const float*)d_in[1];
  const int*   src       = (const int*)d_in[2];
  const float* W         = (const float*)d_in[3];
  float* out = (float*)d_out;

  int n_nodes = in_sizes[0] / D_FEAT;
  int n_edges = in_sizes[2];

  float* aggsum = (float*)d_ws;                                // n_nodes * 128
  float* cnt    = aggsum + (size_t)n_nodes * D_FEAT;           // n_nodes

  long long nz = (long long)n_nodes * D_FEAT + n_nodes;
  zero_f32<<<(int)((nz + 255) / 256), 256, 0, stream>>>(aggsum, nz);

  int blocks1 = (n_edges + 7) / 8;        // 8 edges (waves) per 256-thread block
  edge_scatter<<<blocks1, 256, 0, stream>>>(nbr_feat, src, aggsum, cnt, n_edges);

  int nrb = (n_nodes + 15) / 16;
  int blocks2 = (nrb + 7) / 8;            // 8 row-block waves per block
  size_t lds_bytes = (size_t)OUT_DIM * LDS_PITCH * sizeof(float);  // 130 KB
  fused_mean_gemm<<<blocks2, 256, lds_bytes, stream>>>(
      self_feat, aggsum, cnt, W, out, n_nodes, nrb);
}